// SingleHashNMSKPt2_16338055594733
// MI455X (gfx1250) — compile-verified
//
#include <hip/hip_runtime.h>
#include <cstdint>

// ---------------------------------------------------------------------------
// SingleHashNMS: code hashing -> unique(sorted) -> per-slot argmax(conf)
// N = 8192 rects. The whole working set (64KB int64 codes + 32KB conf) lives
// in one WGP's 320KB LDS on CDNA5. Staging uses BOTH gfx1250 async paths:
//   - TDM tensor_load_to_lds (TENSORcnt) for the 64KB code tile
//   - per-lane global_load_async_to_lds_b32 (ASYNCcnt) for conf
// followed by an in-LDS bitonic sort, unique-rank prefix sum, and a packed
// 64-bit atomic argmax scatter.
// ---------------------------------------------------------------------------

#define NRECTS       8192
#define SORT_THREADS 1024
#define ELEMS_PER_T  (NRECTS / SORT_THREADS)   // 8

typedef unsigned int u32x4 __attribute__((ext_vector_type(4)));
typedef unsigned int u32x8 __attribute__((ext_vector_type(8)));

// ---------------------------------------------------------------------------
// Kernel 1: per-rect hash codes (float32 math mirroring the reference).
// ---------------------------------------------------------------------------
__global__ void k_compute_codes(const float* __restrict__ rects,
                                long long* __restrict__ g_codes)
{
    int i = blockIdx.x * blockDim.x + threadIdx.x;
    if (i >= NRECTS) return;

    // prefetch the next block's rect data (emits global_prefetch_b8)
    if (i + 256 < NRECTS) {
        __builtin_prefetch(rects + (size_t)(i + 256) * 4, 0, 0);
    }

    float x = rects[i * 4 + 0];
    float y = rects[i * 4 + 1];
    float w = rects[i * 4 + 2];
    float h = rects[i * 4 + 3];

    const float log_alpha = logf(1.2f);
    const float log_w0    = logf(512.0f);
    const float log_h0    = logf(512.0f);

    // jnp.round == round-half-to-even -> rintf
    float fi = rintf((log_w0 - logf(w)) / log_alpha);
    float fj = rintf((log_h0 - logf(h)) / log_alpha);

    const float gamma_ratio = 0.25f;               // (1-0.6)/(1+0.6)
    float di = 512.0f * gamma_ratio / powf(1.2f, fi);
    float dj = 512.0f * gamma_ratio / powf(1.2f, fj);

    float qx = rintf(x / di - 0.5f);
    float qy = rintf(y / dj - 0.5f);

    long long code = (long long)qx
                   + (long long)qy * 10000LL
                   + (long long)fi * 100000000LL
                   + (long long)fj * 1000000000000LL;
    g_codes[i] = code;
}

// ---------------------------------------------------------------------------
// Kernel 2: zero the packed argmax slots.
// ---------------------------------------------------------------------------
__global__ void k_init_best(unsigned long long* __restrict__ g_best)
{
    int i = blockIdx.x * blockDim.x + threadIdx.x;
    if (i < NRECTS) g_best[i] = 0ULL;
}

// ---------------------------------------------------------------------------
// Kernel 3: single-workgroup sort + unique-rank + argmax scatter.
//   Dynamic LDS layout:
//     s_codes : 8192 * int64  = 65536 B  (TDM-loaded, sorted in place)
//     s_conf  : 8192 * float  = 32768 B  (async-loaded)
//     s_rank  : 8192 * int    = 32768 B  (unique rank per sorted position)
//     s_part  : 1024 * int    =  4096 B  (prefix-sum partials)
// ---------------------------------------------------------------------------
__global__ void k_sort_rank_argmax(const long long* __restrict__ g_codes,
                                   const float* __restrict__ conf,
                                   unsigned long long* __restrict__ g_best)
{
    extern __shared__ unsigned char smem[];
    long long* s_codes = (long long*)smem;
    float*     s_conf  = (float*)(smem + (size_t)NRECTS * 8);
    int*       s_rank  = (int*)(smem + (size_t)NRECTS * 8 + (size_t)NRECTS * 4);
    int*       s_part  = (int*)(smem + (size_t)NRECTS * 8 + (size_t)NRECTS * 8);

    const int t = threadIdx.x;

    // ---- Stage 0a: per-lane async global->LDS copy of conf (ASYNCcnt path)
    for (int e = 0; e < ELEMS_PER_T; ++e) {
        int idx = t + e * SORT_THREADS;                 // coalesced across lanes
        unsigned long long gaddr = (unsigned long long)(conf + idx);
        unsigned lds_addr = (unsigned)(unsigned long long)(s_conf + idx);
        asm volatile("global_load_async_to_lds_b32 %0, %1, off"
                     :: "v"(lds_addr), "v"(gaddr)
                     : "memory");
    }

    // ---- Stage 0b: TDM descriptor load of the 64KB code tile (TENSORcnt).
    // D# per CDNA5 ISA ch.8: group0 = {count/lds_addr/global_addr/type},
    // group1 = {data_size=8B, tensor_dim0=tile_dim0=8192, 1-D tile}.
    if (t == 0) {
        unsigned long long gaddr = (unsigned long long)g_codes;
        unsigned lds_base = (unsigned)(unsigned long long)s_codes;

        u32x4 d0;
        d0.x = 1u;                                        // count=1, user mode
        d0.y = lds_base;                                  // lds_addr (bytes)
        d0.z = (unsigned)(gaddr & 0xFFFFFFFFu);           // global_addr[31:0]
        d0.w = (unsigned)((gaddr >> 32) & 0x01FFFFFFu)    // global_addr[56:32]
             | (2u << 30);                                // type = 2 ("image")

        u32x8 d1;
        d1.s0 = (3u << 16);                               // data_size=3 (8B), no mask
        d1.s1 = (NRECTS & 0xFFFFu) << 16;                 // tensor_dim0[15:0]
        d1.s2 = ((NRECTS >> 16) & 0xFFFFu)                // tensor_dim0[31:16]
              | (1u << 16);                               // tensor_dim1 = 1
        d1.s3 = (NRECTS & 0xFFFFu) << 16;                 // tile_dim0 = 8192
        d1.s4 = 0u;                                       // tile_dim1/2 unused
        d1.s5 = (unsigned)NRECTS;                         // tensor_dim0_stride[31:0]
        d1.s6 = 0u;                                       // stride hi / dim1_stride lo
        d1.s7 = 0u;                                       // dim1_stride hi

        asm volatile("tensor_load_to_lds %0, %1"
                     :: "s"(d0), "s"(d1)
                     : "memory");
    }

    __builtin_amdgcn_s_wait_tensorcnt(0);                 // s_wait_tensorcnt 0
    asm volatile("s_wait_asynccnt 0" ::: "memory");
    __syncthreads();

    // ---- Stage 1: bitonic sort of 8192 int64 keys in LDS
    for (int k = 2; k <= NRECTS; k <<= 1) {
        for (int j = k >> 1; j > 0; j >>= 1) {
            for (int idx = t; idx < NRECTS; idx += SORT_THREADS) {
                int ixj = idx ^ j;
                if (ixj > idx) {
                    long long a = s_codes[idx];
                    long long b = s_codes[ixj];
                    bool up = ((idx & k) == 0);
                    if ((up && a > b) || (!up && a < b)) {
                        s_codes[idx] = b;
                        s_codes[ixj] = a;
                    }
                }
            }
            __syncthreads();
        }
    }

    // ---- Stage 2: unique ranks via first-occurrence flags + prefix sum
    {
        int base_p = t * ELEMS_PER_T;
        int local_sum = 0;
        for (int e = 0; e < ELEMS_PER_T; ++e) {
            int p = base_p + e;
            int f = (p == 0) || (s_codes[p] != s_codes[p - 1]);
            local_sum += f;
        }
        s_part[t] = local_sum;
        __syncthreads();

        // Hillis-Steele inclusive scan over 1024 partials
        for (int off = 1; off < SORT_THREADS; off <<= 1) {
            int v   = s_part[t];
            int add = (t >= off) ? s_part[t - off] : 0;
            __syncthreads();
            s_part[t] = v + add;
            __syncthreads();
        }

        int running = s_part[t] - local_sum;   // exclusive base for this thread
        for (int e = 0; e < ELEMS_PER_T; ++e) {
            int p = base_p + e;
            int f = (p == 0) || (s_codes[p] != s_codes[p - 1]);
            running += f;
            s_rank[p] = running - 1;           // 0-based unique rank
        }
    }
    __syncthreads();

    // ---- Stage 3: per-rect slot lookup + packed atomic argmax
    // key = (conf_bits << 32) | (0xFFFFFFFF - i): positive-float bit order
    // means max conf wins; ties pick the smallest original index (matching
    // jnp.argmax first-max semantics).
    for (int e = 0; e < ELEMS_PER_T; ++e) {
        int i = t + e * SORT_THREADS;
        long long c = g_codes[i];              // original-order code (global)
        // lower_bound over sorted LDS codes (c is guaranteed present)
        int lo = 0, hi = NRECTS;
        while (lo < hi) {
            int m = (lo + hi) >> 1;
            if (s_codes[m] < c) lo = m + 1; else hi = m;
        }
        int slot = s_rank[lo];
        unsigned cb = __float_as_uint(s_conf[i]);
        unsigned long long key =
            ((unsigned long long)cb << 32) | (unsigned long long)(0xFFFFFFFFu - (unsigned)i);
        atomicMax(&g_best[slot], key);
    }
}

// ---------------------------------------------------------------------------
// Kernel 4: unpack winners (empty / fill slots -> index 0), write as float.
// ---------------------------------------------------------------------------
__global__ void k_finalize(const unsigned long long* __restrict__ g_best,
                           float* __restrict__ out)
{
    int j = blockIdx.x * blockDim.x + threadIdx.x;
    if (j >= NRECTS) return;
    unsigned long long v = g_best[j];
    unsigned idx = (v == 0ULL) ? 0u : (0xFFFFFFFFu - (unsigned)(v & 0xFFFFFFFFULL));
    out[j] = (float)idx;
}

// ---------------------------------------------------------------------------
extern "C" void kernel_launch(void* const* d_in, const int* in_sizes, int n_in,
                              void* d_out, int out_size, void* d_ws, size_t ws_size,
                              hipStream_t stream)
{
    const float* rects = (const float*)d_in[0];   // (8192, 4) float32
    const float* conf  = (const float*)d_in[1];   // (8192,)  float32
    float* out = (float*)d_out;                   // 8192 indices as float

    long long*          g_codes = (long long*)d_ws;
    unsigned long long* g_best  = (unsigned long long*)((char*)d_ws + (size_t)NRECTS * 8);

    dim3 blk(256), grd(NRECTS / 256);
    k_compute_codes<<<grd, blk, 0, stream>>>(rects, g_codes);
    k_init_best<<<grd, blk, 0, stream>>>(g_best);

    size_t smem_bytes = (size_t)NRECTS * 8       // s_codes
                      + (size_t)NRECTS * 4       // s_conf
                      + (size_t)NRECTS * 4       // s_rank
                      + (size_t)SORT_THREADS * 4;// s_part
    k_sort_rank_argmax<<<1, SORT_THREADS, smem_bytes, stream>>>(g_codes, conf, g_best);

    k_finalize<<<grd, blk, 0, stream>>>(g_best, out);
}